// GNN_Layer_13331578487070
// MI455X (gfx1250) — compile-verified
//
#include <hip/hip_runtime.h>

typedef __attribute__((ext_vector_type(2))) float v2f;
typedef __attribute__((ext_vector_type(8))) float v8f;

// D = A(16x4) * B(4x16) + C, f32 WMMA.
// A layout (32-bit A 16x4): lane = m (mod 16); vgpr v, laneHalf h -> k = v + 2h
// B layout (32-bit B 4x16): lane&15 = n;       vgpr v, laneHalf h -> k = v + 2h
// C/D layout: lane&15 = n; value replicated over rows when A rows are equal.
//
// We broadcast the feature vector over A's rows:
//   A[m,k] = f[kBase+k]            (all rows equal)
//   B[k,n] = W[kBase+k, oBase+n]   (natural row-major W, coalesced per half-wave)
// => D[:, n] = sum_k f[kBase+k] * W[kBase+k, oBase+n]  (partial out, all rows equal)
__device__ __forceinline__ void matvec64_wmma(const float* __restrict__ f,
                                              const float* __restrict__ Wm,
                                              int lane, v8f acc[4]) {
    const int lo = lane & 15;
    const int h  = lane >> 4;            // 0: k = 0,1   1: k = 2,3
    const v8f z = {0.f, 0.f, 0.f, 0.f, 0.f, 0.f, 0.f, 0.f};
    acc[0] = z; acc[1] = z; acc[2] = z; acc[3] = z;

#pragma unroll 4
    for (int kb = 0; kb < 16; ++kb) {    // kBase = 4*kb
        const int k0 = 4 * kb + 2 * h;   // this lane-half's first k
        v2f A;
        A.x = f[k0];                     // cached (feat table is tiny, stays in L2)
        A.y = f[k0 + 1];
        const float* __restrict__ r0 = Wm + (size_t)k0 * 64;
#pragma unroll
        for (int t = 0; t < 4; ++t) {    // oBase = 16*t
            v2f B;
            // W streamed once -> non-temporal so it doesn't evict feat from L2
            B.x = __builtin_nontemporal_load(r0 + 16 * t + lo);
            B.y = __builtin_nontemporal_load(r0 + 64 + 16 * t + lo);
            acc[t] = __builtin_amdgcn_wmma_f32_16x16x4_f32(
                false, A, false, B, (short)0, acc[t], false, false);
        }
    }
}

// out[n,:] = h_bias[n,:] + feat[n,:] @ loop_weight[n,:,:]
__global__ void __launch_bounds__(256)
rgcn_node_loop_kernel(const float* __restrict__ feat,
                      const float* __restrict__ loop_w,
                      const float* __restrict__ h_bias,
                      float* __restrict__ out, int n_nodes) {
    const int wave = (int)((blockIdx.x * blockDim.x + threadIdx.x) >> 5);
    if (wave >= n_nodes) return;                     // wave-uniform; EXEC stays full
    const int lane = (int)(threadIdx.x & 31);
    const int lo = lane & 15;
    const int h  = lane >> 4;

    v8f acc[4];
    matvec64_wmma(feat + (size_t)wave * 64, loop_w + (size_t)wave * 4096, lane, acc);

    // lanes 0-15 own tiles 0,1 (o = lo, lo+16); lanes 16-31 own tiles 2,3 (o = 32+lo, 48+lo)
    const int o1 = h * 32 + lo;
    const float v1 = h ? acc[2][0] : acc[0][0];
    const float v2 = h ? acc[3][0] : acc[1][0];
    const size_t base = (size_t)wave * 64;
    out[base + o1]      = v1 + h_bias[base + o1];
    out[base + o1 + 16] = v2 + h_bias[base + o1 + 16];
}

// out[dst[e],:] += feat[src[e],:] @ W[e,:,:] + m_bias[e,:]
__global__ void __launch_bounds__(256)
rgcn_edge_msg_kernel(const float* __restrict__ feat,
                     const int* __restrict__ src,
                     const int* __restrict__ dst,
                     const float* __restrict__ W,
                     const float* __restrict__ m_bias,
                     float* __restrict__ out, int n_edges) {
    const int e = (int)((blockIdx.x * blockDim.x + threadIdx.x) >> 5);
    if (e >= n_edges) return;                        // wave-uniform; EXEC stays full
    const int lane = (int)(threadIdx.x & 31);
    const int lo = lane & 15;
    const int h  = lane >> 4;

    const int s = src[e];
    const int d = dst[e];

    v8f acc[4];
    matvec64_wmma(feat + (size_t)s * 64, W + (size_t)e * 4096, lane, acc);

    const int o1 = h * 32 + lo;
    const size_t eb = (size_t)e * 64;
    const float v1 = (h ? acc[2][0] : acc[0][0]) +
                     __builtin_nontemporal_load(m_bias + eb + o1);
    const float v2 = (h ? acc[3][0] : acc[1][0]) +
                     __builtin_nontemporal_load(m_bias + eb + o1 + 16);

    float* p = out + (size_t)d * 64;
    // relaxed, device-scope, result unused -> non-returning global_atomic_add_f32
    __hip_atomic_fetch_add(p + o1,      v1, __ATOMIC_RELAXED, __HIP_MEMORY_SCOPE_AGENT);
    __hip_atomic_fetch_add(p + o1 + 16, v2, __ATOMIC_RELAXED, __HIP_MEMORY_SCOPE_AGENT);
}

extern "C" void kernel_launch(void* const* d_in, const int* in_sizes, int n_in,
                              void* d_out, int out_size, void* d_ws, size_t ws_size,
                              hipStream_t stream) {
    const float* feat   = (const float*)d_in[0];
    const int*   src    = (const int*)  d_in[1];
    const int*   dst    = (const int*)  d_in[2];
    const float* W      = (const float*)d_in[3];
    const float* loop_w = (const float*)d_in[4];
    const float* m_bias = (const float*)d_in[5];
    const float* h_bias = (const float*)d_in[6];
    float* out = (float*)d_out;

    const int n_nodes = in_sizes[0] / 64;   // 10000
    const int n_edges = in_sizes[1];        // 100000

    const int threads = 256;                // 8 wave32 per block
    const int wpb = threads / 32;

    // 1) init out with self-loop term + h_bias (plain stores)
    rgcn_node_loop_kernel<<<(n_nodes + wpb - 1) / wpb, threads, 0, stream>>>(
        feat, loop_w, h_bias, out, n_nodes);

    // 2) per-edge messages, scatter-added with f32 atomics (stream-ordered after init)
    rgcn_edge_msg_kernel<<<(n_edges + wpb - 1) / wpb, threads, 0, stream>>>(
        feat, src, dst, W, m_bias, out, n_edges);
}